// GANTacotron_26671746908852
// MI455X (gfx1250) — compile-verified
//
#include <hip/hip_runtime.h>
#include <hip/hip_bf16.h>
#include <stdint.h>

#define B      16
#define C_MEL  80
#define D_RES  256
#define TX     1024
#define TY     4096
#define NEGBIG (-1e9f)

typedef float v2f __attribute__((ext_vector_type(2)));
typedef float v4f __attribute__((ext_vector_type(4)));
typedef float v8f __attribute__((ext_vector_type(8)));

// ---------- K1: squared norms: out[b*T+t] = sum_c src[b][c][t]^2 ----------
__global__ void k_norms(const float* __restrict__ src, float* __restrict__ out, int T) {
  int i = blockIdx.x * blockDim.x + threadIdx.x;
  if (i >= B * T) return;
  int b = i / T, t = i - b * T;
  const float* p = src + (size_t)b * C_MEL * T + t;
  float s = 0.f;
#pragma unroll 8
  for (int c = 0; c < C_MEL; ++c) { float v = p[(size_t)c * T]; s += v * v; }
  out[i] = s;
}

// ---------- K2: corr_t[b][s][t] = 2*(x_m^T mel)[t,s] - |x_m[t]|^2 - |mel[s]|^2 ----------
// WMMA f32 16x16x4, M = ty-tile (rows of A = mel columns), N = tx-tile (cols of B = x_m).
// One 16x16 output tile per wave, K-loop over C_MEL = 80 (20 WMMA steps).
__global__ __launch_bounds__(256) void k_corr_wmma(
    const float* __restrict__ x_m, const float* __restrict__ mel,
    const float* __restrict__ xn, const float* __restrict__ zn,
    float* __restrict__ corr_t) {
  const int lane = threadIdx.x & 31;
  const int wave = threadIdx.x >> 5;
  const int tTiles = TX / 16, sTiles = TY / 16;
  int tile = blockIdx.x * 8 + wave;
  int t0 = (tile % tTiles) * 16;
  int s0 = ((tile / tTiles) % sTiles) * 16;
  int b  = tile / (tTiles * sTiles);
  const int half = lane >> 4, lr = lane & 15;

  // A (16x4, M x K): lanes 0-15 row m=lr {K=k0,k0+1}; lanes 16-31 {K=k0+2,k0+3}
  const float* A  = mel + (size_t)b * C_MEL * TY + s0 + lr;  // A[m][k] = mel[b][k][s0+m]
  const float* Bm = x_m + (size_t)b * C_MEL * TX + t0 + lr;  // B[k][n] = x_m[b][k][t0+n]
  v8f acc = {};
#pragma unroll
  for (int k0 = 0; k0 < C_MEL; k0 += 4) {
    int ka = k0 + 2 * half;
    v2f av, bv;
    av.x = A[(size_t)ka * TY];
    av.y = A[(size_t)(ka + 1) * TY];
    bv.x = Bm[(size_t)ka * TX];
    bv.y = Bm[(size_t)(ka + 1) * TX];
    acc = __builtin_amdgcn_wmma_f32_16x16x4_f32(false, av, false, bv, (short)0, acc,
                                                false, false);
  }
  // D layout: vgpr r, lane l -> M = r + 8*(l>=16), N = l%16
  float xnv = xn[b * TX + t0 + lr];
  const float* znp = zn + b * TY + s0 + 8 * half;
  float* outp = corr_t + ((size_t)b * TY + s0 + 8 * half) * TX + t0 + lr;
#pragma unroll
  for (int r = 0; r < 8; ++r) {
    __builtin_nontemporal_store(2.f * acc[r] - xnv - znp[r], outp + (size_t)r * TX);
  }
}

// ---------- K3: Viterbi forward DP, one workgroup (32 waves) per batch ----------
// v_new[x] = (x<=j) ? max(v[x], v[x-1]) + mask*corr : NEG ; dir = mask ? (v[x]>=v[x-1]) : 1
__global__ __launch_bounds__(1024) void k_dp_forward(
    const float* __restrict__ corr_t, const int* __restrict__ xlen_,
    const int* __restrict__ ylen_, uint32_t* __restrict__ dirs) {
  const int b = blockIdx.x, x = threadIdx.x;
  __shared__ float vb[2][TX + 1];           // [*][0] = NEG sentinel (v[-1])
  const int xlen = xlen_[b], ylen = ylen_[b];
  if (x == 0) { vb[0][0] = NEGBIG; vb[1][0] = NEGBIG; }
  vb[0][x + 1] = 0.f;
  __syncthreads();
  const float* cb = corr_t + (size_t)b * TY * TX + x;
  uint32_t* db = dirs + (size_t)b * TY * (TX / 32) + (x >> 5);
  const bool xin = x < xlen;
  float pipe[4];                            // 4-deep column prefetch
#pragma unroll
  for (int d = 0; d < 4; ++d) pipe[d] = __builtin_nontemporal_load(cb + (size_t)d * TX);
  int p = 0;
  for (int j = 0; j < TY; ++j) {
    float val = pipe[j & 3];
    if (j + 4 < TY) pipe[j & 3] = __builtin_nontemporal_load(cb + (size_t)(j + 4) * TX);
    float vc = vb[p][x + 1];
    float vp = vb[p][x];
    bool stay = vc >= vp;
    bool inmask = xin && (j < ylen);
    float vmax = stay ? vc : vp;
    float vnew = (x <= j) ? (vmax + (inmask ? val : 0.f)) : NEGBIG;
    uint32_t word = __builtin_amdgcn_ballot_w32(inmask ? stay : true);
    if ((x & 31) == 0) db[(size_t)j * (TX / 32)] = word;
    vb[p ^ 1][x + 1] = vnew;
    __syncthreads();
    p ^= 1;
  }
}

// ---------- K4: backtrace, one wave per batch ----------
// 32 steps per chunk: idx moves <=1/step so all dir bits live in <=2 words per column;
// wave loads 32 columns' words, then walks 32 steps in registers via ds_bpermute.
__global__ __launch_bounds__(32) void k_backtrace(
    const uint32_t* __restrict__ dirs, const int* __restrict__ xlen_,
    const int* __restrict__ ylen_, int* __restrict__ idx_out,
    int* __restrict__ counts) {
  const int b = blockIdx.x, lane = threadIdx.x;
  __shared__ int cnt[TX];
  for (int x = lane; x < TX; x += 32) cnt[x] = 0;
  __syncthreads();
  const int xlen = xlen_[b], ylen = ylen_[b];
  const uint32_t* db = dirs + (size_t)b * TY * (TX / 32);
  int idx = xlen - 1;
  for (int hi = TY - 1; hi >= 0; hi -= 32) {
    const int nsteps = (hi + 1 < 32) ? (hi + 1) : 32;
    const int wlo = ((idx > 31 ? (idx - 31) : 0)) >> 5;
    uint32_t w0 = 0xFFFFFFFFu, w1 = 0xFFFFFFFFu;
    if (lane < nsteps) {
      const uint32_t* col = db + (size_t)(hi - lane) * (TX / 32);
      w0 = col[wlo];
      w1 = (wlo + 1 < TX / 32) ? col[wlo + 1] : 0xFFFFFFFFu;
    }
    int my_idx = 0;
    for (int s = 0; s < nsteps; ++s) {      // uniform serial walk, j = hi - s
      if (lane == s) my_idx = idx;          // one-hot row recorded pre-update
      int wsel = (idx >> 5) - wlo;          // uniform selector (0 or 1)
      int wp = (int)(wsel == 0 ? w0 : w1);
      uint32_t w = (uint32_t)__builtin_amdgcn_ds_bpermute(s << 2, wp);
      uint32_t bit = (w >> (idx & 31)) & 1u;
      if ((hi - s) < ylen && lane == 0) cnt[idx]++;
      idx += (int)bit - 1;                  // dir==1 stay, dir==0 step down
      if (idx < 0) idx = 0;
    }
    if (lane < nsteps) idx_out[b * TY + (hi - lane)] = my_idx;
  }
  __syncthreads();
  for (int x = lane; x < TX; x += 32) counts[b * TX + x] = cnt[x];
}

// ---------- K5a: dense one-hot attn (256 MB streaming write) ----------
__global__ void k_attn(const int* __restrict__ idx_arr, const int* __restrict__ ylen_,
                       v4f* __restrict__ attn) {
  size_t i = (size_t)blockIdx.x * blockDim.x + threadIdx.x;  // B*TX*(TY/4)
  const int S4 = TY / 4;
  int s4 = (int)(i % S4);
  int x  = (int)((i / S4) % TX);
  int b  = (int)(i / ((size_t)S4 * TX));
  int ylen = ylen_[b];
  const int* ip = idx_arr + b * TY + s4 * 4;
  v4f o;
  o.x = (s4 * 4 + 0 < ylen && ip[0] == x) ? 1.f : 0.f;
  o.y = (s4 * 4 + 1 < ylen && ip[1] == x) ? 1.f : 0.f;
  o.z = (s4 * 4 + 2 < ylen && ip[2] == x) ? 1.f : 0.f;
  o.w = (s4 * 4 + 3 < ylen && ip[3] == x) ? 1.f : 0.f;
  __builtin_nontemporal_store(o, attn + i);
}

// ---------- K5b: z_m = gather of x_res along the path ----------
__global__ void k_zm(const float* __restrict__ x_res, const int* __restrict__ idx_arr,
                     const int* __restrict__ ylen_, float* __restrict__ z_m) {
  size_t i = (size_t)blockIdx.x * blockDim.x + threadIdx.x;  // B*D_RES*TY
  int s = (int)(i % TY);
  int c = (int)((i / TY) % D_RES);
  int b = (int)(i / ((size_t)TY * D_RES));
  float v = 0.f;
  if (s < ylen_[b]) {
    int xi = idx_arr[b * TY + s];
    v = x_res[((size_t)b * D_RES + c) * TX + xi];
  }
  __builtin_nontemporal_store(v, z_m + i);
}

// ---------- K5c: mel_proj = gather(x_m) + noise; masked L1 loss partial ----------
__global__ __launch_bounds__(256) void k_melproj(
    const float* __restrict__ x_m, const float* __restrict__ mel,
    const float* __restrict__ noise, const int* __restrict__ idx_arr,
    const int* __restrict__ ylen_, float* __restrict__ mel_proj,
    float* __restrict__ loss_acc) {
  __shared__ float red[256];
  size_t i = (size_t)blockIdx.x * blockDim.x + threadIdx.x;  // B*C_MEL*TY
  int s = (int)(i % TY);
  int c = (int)((i / TY) % C_MEL);
  int b = (int)(i / ((size_t)TY * C_MEL));
  int ylen = ylen_[b];
  float g = 0.f;
  if (s < ylen) {
    int xi = idx_arr[b * TY + s];
    g = x_m[((size_t)b * C_MEL + c) * TX + xi];
  }
  float mp = g + noise[i];
  __builtin_nontemporal_store(mp, mel_proj + i);
  red[threadIdx.x] = (s < ylen) ? fabsf(mp - mel[i]) : 0.f;
  __syncthreads();
  for (int off = 128; off > 0; off >>= 1) {
    if (threadIdx.x < off) red[threadIdx.x] += red[threadIdx.x + off];
    __syncthreads();
  }
  if (threadIdx.x == 0) atomicAdd(loss_acc, red[0]);
}

// ---------- K5d: z_mask + logw ----------
__global__ void k_small(const int* __restrict__ xlen_, const int* __restrict__ ylen_,
                        const int* __restrict__ counts, float* __restrict__ z_mask,
                        float* __restrict__ logw) {
  int i = blockIdx.x * blockDim.x + threadIdx.x;
  if (i < B * TY) {
    int b = i / TY, s = i - b * TY;
    z_mask[i] = (s < ylen_[b]) ? 1.f : 0.f;
  }
  if (i < B * TX) {
    int b = i / TX, x = i - b * TX;
    logw[i] = (x < xlen_[b]) ? logf(1e-8f + (float)counts[i]) : 0.f;
  }
}

// ---------- K5e: finalize loss mean ----------
__global__ void k_loss_fin(const float* __restrict__ loss_acc, float* __restrict__ out) {
  out[0] = loss_acc[0] / (float)(B * C_MEL * TY);
}

extern "C" void kernel_launch(void* const* d_in, const int* in_sizes, int n_in,
                              void* d_out, int out_size, void* d_ws, size_t ws_size,
                              hipStream_t stream) {
  (void)in_sizes; (void)n_in; (void)out_size; (void)ws_size;
  const float* x_m   = (const float*)d_in[0];
  const float* x_res = (const float*)d_in[1];
  const float* mel   = (const float*)d_in[2];
  const int*   xlen  = (const int*)d_in[3];
  const int*   ylen  = (const int*)d_in[4];
  const float* noise = (const float*)d_in[5];

  // workspace layout
  char* ws = (char*)d_ws;
  float*    corr_t = (float*)ws;                                 // 256 MB [B][TY][TX]
  uint32_t* dirs   = (uint32_t*)(ws + (size_t)B * TY * TX * 4);  // 8 MB  [B][TY][TX/32]
  char* p = ws + (size_t)B * TY * TX * 4 + (size_t)B * TY * (TX / 32) * 4;
  float* xn       = (float*)p; p += (size_t)B * TX * 4;
  float* zn       = (float*)p; p += (size_t)B * TY * 4;
  int*   idx_arr  = (int*)p;   p += (size_t)B * TY * 4;
  int*   counts   = (int*)p;   p += (size_t)B * TX * 4;
  float* loss_acc = (float*)p;

  // output layout (flat concat in reference return order)
  float* out     = (float*)d_out;
  float* o_zm    = out;                                   // [B][D_RES][TY]
  float* o_zmask = o_zm + (size_t)B * D_RES * TY;         // [B][1][TY]
  float* o_attn  = o_zmask + (size_t)B * TY;              // [B][TX][TY]
  float* o_logw  = o_attn + (size_t)B * TX * TY;          // [B][1][TX]
  float* o_loss  = o_logw + (size_t)B * TX;               // scalar
  float* o_mp    = o_loss + 1;                            // [B][C_MEL][TY]

  (void)hipMemsetAsync(loss_acc, 0, sizeof(float), stream);

  k_norms<<<(B * TX + 255) / 256, 256, 0, stream>>>(x_m, xn, TX);
  k_norms<<<(B * TY + 255) / 256, 256, 0, stream>>>(mel, zn, TY);
  k_corr_wmma<<<(B * (TY / 16) * (TX / 16)) / 8, 256, 0, stream>>>(x_m, mel, xn, zn, corr_t);
  k_dp_forward<<<B, 1024, 0, stream>>>(corr_t, xlen, ylen, dirs);
  k_backtrace<<<B, 32, 0, stream>>>(dirs, xlen, ylen, idx_arr, counts);
  k_attn<<<(int)(((size_t)B * TX * (TY / 4)) / 256), 256, 0, stream>>>(idx_arr, ylen,
                                                                       (v4f*)o_attn);
  k_zm<<<(int)(((size_t)B * D_RES * TY) / 256), 256, 0, stream>>>(x_res, idx_arr, ylen, o_zm);
  k_melproj<<<(int)(((size_t)B * C_MEL * TY) / 256), 256, 0, stream>>>(x_m, mel, noise,
                                                                       idx_arr, ylen, o_mp,
                                                                       loss_acc);
  k_small<<<(B * TY + 255) / 256, 256, 0, stream>>>(xlen, ylen, counts, o_zmask, o_logw);
  k_loss_fin<<<1, 1, 0, stream>>>(loss_acc, o_loss);
}